// GraphAttentionLayer_52905407152165
// MI455X (gfx1250) — compile-verified
//
#include <hip/hip_runtime.h>
#include <hip/hip_bf16.h>

// ---------------------------------------------------------------------------
// GAT layer for MI455X (gfx1250, wave32).
// Both GEMMs (X·W and Attn·h) use the CDNA5 fp32 matrix instruction
// V_WMMA_F32_16X16X4_F32 to keep full fp32 reference precision.
// Problem is tiny (~0.35 GFLOP, ~15 MB) => L2-resident; LDS-staged tiles.
// ---------------------------------------------------------------------------

typedef __attribute__((ext_vector_type(2))) float v2f;
typedef __attribute__((ext_vector_type(8))) float v8f;

#define BATCH   32
#define N_NODES 228
#define FIN     128
#define FOUT    64
#define ROW_TILES ((N_NODES + 15) / 16)   // 15
#define NEG_INF_F (-9.0e15f)
#define GAT_ALPHA 0.2f
#define BN_EPS    1e-5f

// ---------------------------------------------------------------------------
// Kernel 1: h = input @ W   (per block: one batch b, one 16-row tile of N)
// 128 threads = 4 waves; wave w computes output cols [16w, 16w+16).
// ---------------------------------------------------------------------------
__global__ __launch_bounds__(128) void gat_gemm_h(
    const float* __restrict__ inp,   // (B, N, FIN)
    const float* __restrict__ W,     // (FIN, FOUT)
    float* __restrict__ h)           // (B, N, FOUT)
{
    __shared__ __align__(16) float sW[FIN * FOUT];   // 32 KB
    __shared__ __align__(16) float sA[16 * FIN];     // 8 KB

    const int tid = threadIdx.x;
    const int b   = blockIdx.x / ROW_TILES;
    const int i0  = (blockIdx.x % ROW_TILES) * 16;

    // stage W (full) into LDS
    for (int idx = tid; idx < FIN * FOUT / 4; idx += 128)
        ((float4*)sW)[idx] = ((const float4*)W)[idx];

    // stage 16xFIN input tile (clamp OOB rows; their results are never stored)
    for (int idx = tid; idx < 16 * FIN / 4; idx += 128) {
        int r   = idx / (FIN / 4);
        int c4  = idx % (FIN / 4);
        int row = i0 + r; if (row > N_NODES - 1) row = N_NODES - 1;
        ((float4*)sA)[idx] =
            ((const float4*)(inp + ((size_t)b * N_NODES + row) * FIN))[c4];
    }
    __syncthreads();

    const int wave = tid >> 5;
    const int lane = tid & 31;
    const int hi   = lane >> 4;     // 0: K={k,k+1} half, 1: K={k+2,k+3} half
    const int l16  = lane & 15;
    const int n0   = wave * 16;

    v8f acc = {};
    for (int k = 0; k < FIN; k += 4) {
        const int kk = k + hi * 2;
        // A frag: row=l16, contiguous K pair -> single ds_load_b64
        v2f a2 = *(const v2f*)(sA + l16 * FIN + kk);
        // B frag: W[kk][n0+l16], W[kk+1][n0+l16]
        v2f b2;
        b2.x = sW[kk * FOUT + n0 + l16];
        b2.y = sW[(kk + 1) * FOUT + n0 + l16];
        acc = __builtin_amdgcn_wmma_f32_16x16x4_f32(
            false, a2, false, b2, (short)0, acc, false, false);
    }

    // D layout: VGPR r -> M = r (lanes 0-15) / r+8 (lanes 16-31), N = l16
    for (int r = 0; r < 8; ++r) {
        int row = i0 + r + hi * 8;
        if (row < N_NODES)
            h[((size_t)b * N_NODES + row) * FOUT + n0 + l16] = acc[r];
    }
}

// ---------------------------------------------------------------------------
// Kernel 2: esrc[b,i] = h[b,i,:]·a[:64],  edst[b,i] = h[b,i,:]·a[64:128]
// ---------------------------------------------------------------------------
__global__ __launch_bounds__(256) void gat_attn_vec(
    const float* __restrict__ h, const float* __restrict__ a,
    float* __restrict__ esrc, float* __restrict__ edst)
{
    int idx = blockIdx.x * 256 + threadIdx.x;      // b*N + i
    if (idx >= BATCH * N_NODES) return;
    const float* hr = h + (size_t)idx * FOUT;
    float s0 = 0.f, s1 = 0.f;
    #pragma unroll 8
    for (int f = 0; f < FOUT; ++f) {
        float v = hr[f];
        s0 += v * a[f];
        s1 += v * a[FOUT + f];
    }
    esrc[idx] = s0;
    edst[idx] = s1;
}

// ---------------------------------------------------------------------------
// Kernel 3: masked LeakyReLU + softmax over j, then h' = Attn @ h, then ELU.
// One block per (b, 16-row tile). Scores/probs live in a 16x240 LDS tile
// (stride 240 keeps ds_load_b64 8B-aligned). WMMA K-loop runs j=0..227.
// ---------------------------------------------------------------------------
__global__ __launch_bounds__(128) void gat_softmax_aggr(
    const float* __restrict__ h,    // (B, N, FOUT)
    const int*   __restrict__ adj,  // (B, N, N)
    const float* __restrict__ esrc,
    const float* __restrict__ edst,
    float* __restrict__ hp)         // (B, N, FOUT)
{
    __shared__ __align__(16) float sP[16 * 240];   // 15.4 KB score/prob tile
    __shared__ float sEdst[N_NODES];
    __shared__ float sRed[16][8];
    __shared__ float sRowMax[16];
    __shared__ float sInvSum[16];

    const int tid = threadIdx.x;
    const int b   = blockIdx.x / ROW_TILES;
    const int i0  = (blockIdx.x % ROW_TILES) * 16;

    for (int j = tid; j < N_NODES; j += 128)
        sEdst[j] = edst[b * N_NODES + j];

    // h[b] is L2-resident (1.9 MB << 192 MB L2): prefetch it (global_prefetch_b8)
    {
        const char* base = (const char*)(h + (size_t)b * N_NODES * FOUT);
        for (int off = tid * 128; off < N_NODES * FOUT * 4; off += 128 * 128)
            __builtin_prefetch(base + off, 0, 3);
    }
    __syncthreads();

    const int sub = tid & 7;   // 8 threads cooperate on one row
    const int r   = tid >> 3;  // local row 0..15
    const int i   = i0 + r;
    const bool rowValid = (i < N_NODES);
    const float es = rowValid ? esrc[b * N_NODES + i] : 0.0f;
    const int* adjRow = adj + ((size_t)b * N_NODES + (rowValid ? i : 0)) * N_NODES;

    // pass 1: masked LeakyReLU scores + row max
    float lmax = NEG_INF_F;
    for (int j = sub; j < N_NODES; j += 8) {
        float s;
        if (rowValid) {
            float e = es + sEdst[j];
            e = (e > 0.0f) ? e : GAT_ALPHA * e;
            s = (adjRow[j] > 0) ? e : NEG_INF_F;
        } else {
            s = 0.0f;   // dummy rows: finite scores, results never stored
        }
        sP[r * 240 + j] = s;
        lmax = fmaxf(lmax, s);
    }
    sRed[r][sub] = lmax;
    __syncthreads();
    if (sub == 0) {
        float m = sRed[r][0];
        for (int t = 1; t < 8; ++t) m = fmaxf(m, sRed[r][t]);
        sRowMax[r] = m;
    }
    __syncthreads();

    // pass 2: exp(s - rowmax) + row sum   (all-masked row -> uniform, like jax)
    const float rm = sRowMax[r];
    float lsum = 0.0f;
    for (int j = sub; j < N_NODES; j += 8) {
        float p = __expf(sP[r * 240 + j] - rm);
        sP[r * 240 + j] = p;
        lsum += p;
    }
    sRed[r][sub] = lsum;
    __syncthreads();
    if (sub == 0) {
        float ssum = 0.0f;
        for (int t = 0; t < 8; ++t) ssum += sRed[r][t];
        sInvSum[r] = 1.0f / ssum;
    }
    __syncthreads();

    // pass 3: h'(16x64) = P(16x228) @ h[b](228x64) via fp32 WMMA, K step 4.
    // 228 % 4 == 0 -> no tail, no OOB, EXEC uniform through every WMMA.
    const int wave = tid >> 5;
    const int lane = tid & 31;
    const int hi   = lane >> 4;
    const int l16  = lane & 15;
    const int n0   = wave * 16;
    const float* hb = h + (size_t)b * N_NODES * FOUT;

    v8f acc = {};
    for (int k = 0; k < N_NODES; k += 4) {
        const int kk = k + hi * 2;
        v2f a2 = *(const v2f*)(sP + l16 * 240 + kk);     // ds_load_b64
        v2f b2;
        b2.x = hb[(size_t)kk * FOUT + n0 + l16];
        b2.y = hb[(size_t)(kk + 1) * FOUT + n0 + l16];
        acc = __builtin_amdgcn_wmma_f32_16x16x4_f32(
            false, a2, false, b2, (short)0, acc, false, false);
    }

    // normalize by row sum, ELU, store h'
    for (int rr = 0; rr < 8; ++rr) {
        int row = i0 + rr + hi * 8;
        if (row < N_NODES) {
            float v = acc[rr] * sInvSum[rr + hi * 8];
            v = (v > 0.0f) ? v : (__expf(v) - 1.0f);   // jax.nn.elu
            hp[((size_t)b * N_NODES + row) * FOUT + n0 + l16] = v;
        }
    }
}

// ---------------------------------------------------------------------------
// Kernel 4: per-node batch norm over (B, FOUT) = 2048 values; out = g*xhat + b
// ---------------------------------------------------------------------------
__global__ __launch_bounds__(256) void gat_bn(
    const float* __restrict__ hp, const float* __restrict__ gamma,
    const float* __restrict__ beta, float* __restrict__ out)
{
    const int n   = blockIdx.x;
    const int tid = threadIdx.x;
    __shared__ float sSum[256];
    __shared__ float sSq[256];

    float vals[8];
    float s = 0.f, sq = 0.f;
    #pragma unroll
    for (int t = 0; t < 8; ++t) {
        int idx = tid + t * 256;           // 0..2047
        int b = idx >> 6, f = idx & 63;
        float v = hp[((size_t)b * N_NODES + n) * FOUT + f];
        vals[t] = v; s += v; sq += v * v;
    }
    sSum[tid] = s; sSq[tid] = sq;
    __syncthreads();
    for (int off = 128; off > 0; off >>= 1) {
        if (tid < off) { sSum[tid] += sSum[tid + off]; sSq[tid] += sSq[tid + off]; }
        __syncthreads();
    }
    const float mean = sSum[0] * (1.0f / 2048.0f);
    const float var  = sSq[0] * (1.0f / 2048.0f) - mean * mean;  // jnp.var ddof=0
    const float rstd = rsqrtf(var + BN_EPS);
    const float g = gamma[n], be = beta[n];
    #pragma unroll
    for (int t = 0; t < 8; ++t) {
        int idx = tid + t * 256;
        int b = idx >> 6, f = idx & 63;
        out[((size_t)b * N_NODES + n) * FOUT + f] = g * (vals[t] - mean) * rstd + be;
    }
}

// ---------------------------------------------------------------------------
extern "C" void kernel_launch(void* const* d_in, const int* in_sizes, int n_in,
                              void* d_out, int out_size, void* d_ws, size_t ws_size,
                              hipStream_t stream) {
    const float* inp   = (const float*)d_in[0];   // (32, 228, 128) f32
    const int*   adj   = (const int*)  d_in[1];   // (32, 228, 228) i32
    const float* W     = (const float*)d_in[2];   // (128, 64) f32
    const float* a     = (const float*)d_in[3];   // (128, 1) f32
    const float* gamma = (const float*)d_in[4];   // (228,) f32
    const float* beta  = (const float*)d_in[5];   // (228,) f32
    float* out = (float*)d_out;                   // (32, 228, 64) f32

    float* ws   = (float*)d_ws;
    float* h    = ws;                                   // 466944 floats
    float* hp   = h + (size_t)BATCH * N_NODES * FOUT;   // 466944 floats
    float* esrc = hp + (size_t)BATCH * N_NODES * FOUT;  // 7296 floats
    float* edst = esrc + BATCH * N_NODES;               // 7296 floats

    gat_gemm_h<<<BATCH * ROW_TILES, 128, 0, stream>>>(inp, W, h);
    gat_attn_vec<<<(BATCH * N_NODES + 255) / 256, 256, 0, stream>>>(h, a, esrc, edst);
    gat_softmax_aggr<<<BATCH * ROW_TILES, 128, 0, stream>>>(h, adj, esrc, edst, hp);
    gat_bn<<<N_NODES, 256, 0, stream>>>(hp, gamma, beta, out);
}